// Model_29145648071115
// MI455X (gfx1250) — compile-verified
//
#include <hip/hip_runtime.h>
#include <hip/hip_bf16.h>

// Sizes from the reference
#define BATCH   2048
#define RULES   1024
#define ATT     64
#define RES     8

typedef float v2f __attribute__((ext_vector_type(2)));
typedef float v4f __attribute__((ext_vector_type(4)));
typedef float v8f __attribute__((ext_vector_type(8)));

// ---------------------------------------------------------------------------
// Prep 1: att_w = softmax(E)   (64 values, 1 block of 64 threads)
// ---------------------------------------------------------------------------
__global__ __launch_bounds__(64) void prep_attw(const float* __restrict__ E,
                                                float* __restrict__ attw) {
    __shared__ float s[ATT];
    int t = threadIdx.x;
    float v = E[t];
    s[t] = v;
    __syncthreads();
    float mx = -3.4e38f;
    #pragma unroll
    for (int i = 0; i < ATT; ++i) mx = fmaxf(mx, s[i]);
    float ex = expf(v - mx);
    __syncthreads();
    s[t] = ex;
    __syncthreads();
    float sum = 0.f;
    #pragma unroll
    for (int i = 0; i < ATT; ++i) sum += s[i];
    attw[t] = ex / sum;
}

// ---------------------------------------------------------------------------
// Prep 2: per-rule A2[r] = sum_a att_w[a]*A[r,a]^2 ; Rsq = R^2 ;
//         beta[r,:] = softmax(B[r,:])
// ---------------------------------------------------------------------------
__global__ __launch_bounds__(128) void prep_rules(
    const float* __restrict__ Ar, const float* __restrict__ B,
    const float* __restrict__ R,  const float* __restrict__ attw,
    float* __restrict__ A2, float* __restrict__ Rsq, float* __restrict__ beta) {
    __shared__ float sw[ATT];
    int tid = threadIdx.x;
    if (tid < ATT) sw[tid] = attw[tid];
    __syncthreads();
    int r = blockIdx.x * 128 + tid;
    const float* arow = Ar + r * ATT;
    float s = 0.f;
    #pragma unroll 8
    for (int a = 0; a < ATT; ++a) { float av = arow[a]; s = fmaf(sw[a] * av, av, s); }
    A2[r] = s;
    float rr = R[r];
    Rsq[r] = rr * rr;
    float bl[RES];
    float mx = -3.4e38f;
    #pragma unroll
    for (int k = 0; k < RES; ++k) { bl[k] = B[r * RES + k]; mx = fmaxf(mx, bl[k]); }
    float se = 0.f;
    #pragma unroll
    for (int k = 0; k < RES; ++k) { bl[k] = expf(bl[k] - mx); se += bl[k]; }
    float inv = 1.f / se;
    #pragma unroll
    for (int k = 0; k < RES; ++k) beta[r * RES + k] = bl[k] * inv;
}

// ---------------------------------------------------------------------------
// Main: one block per 16 batch rows. 8 waves; each wave does 8 rule tiles of
// 16 via chained V_WMMA_F32_16X16X4_F32 (K=64 -> 16 MACC steps, 4-way split
// accumulators). Fully unrolled so tile t+1 loads overlap tile t math.
// ---------------------------------------------------------------------------
__global__ __launch_bounds__(256) void brb_main(
    const float* __restrict__ x,    const float* __restrict__ Ar,
    const float* __restrict__ O,    const float* __restrict__ U,
    const float* __restrict__ attw, const float* __restrict__ A2,
    const float* __restrict__ Rsq,  const float* __restrict__ beta,
    float* __restrict__ out) {
    extern __shared__ float smem[];
    float* lds_e    = smem;                    // 16 * 1024
    float* lds_beta = lds_e + 16 * RULES;      // 1024 * 8
    float* lds_part = lds_beta + RULES * RES;  // 16 * 16
    float* lds_inv  = lds_part + 256;          // 16
    float* lds_p    = lds_inv + 16;            // 16 * 9

    const int tid  = threadIdx.x;
    const int wave = tid >> 5;                 // wave32
    const int lane = tid & 31;
    const int hi   = lane >> 4;                // upper half-wave
    const int ln   = lane & 15;
    const int b0   = blockIdx.x * 16;

    // Cooperative copy of beta (32 KB) into LDS for the ER-product stage.
    {
        const v4f* src = (const v4f*)beta;
        v4f*       dst = (v4f*)lds_beta;
        #pragma unroll
        for (int i = 0; i < (RULES * RES / 4) / 256; ++i)
            dst[tid + 256 * i] = src[tid + 256 * i];
    }

    const float invO2 = 1.0f / (O[0] * O[0]);

    // --- A-operand fragments: x-tile (16 x 64), scaled by att_w -----------
    // f32 16x16x4 A layout: lanes 0-15 M=lane, K={4c,4c+1}; lanes 16-31
    // M=lane-16, K={4c+2,4c+3}.
    v2f afrag[16];
    {
        const float* xrow = x + (size_t)(b0 + ln) * ATT;
        #pragma unroll
        for (int c = 0; c < 16; ++c) {
            int k = 4 * c + 2 * hi;
            afrag[c].x = xrow[k]     * attw[k];
            afrag[c].y = xrow[k + 1] * attw[k + 1];
        }
    }

    // --- 8 rule tiles per wave (fully unrolled; loads decoupled from MACCs)
    #pragma unroll
    for (int t = 0; t < 8; ++t) {
        const int r0 = (wave * 8 + t) * 16;
        const int r  = r0 + ln;
        const float* arow = Ar + (size_t)r * ATT;
        if (t < 7) __builtin_prefetch(arow + 16 * ATT, 0, 3);

        // Preload all 16 B fragments so the loads issue as one clause.
        v2f bfrag[16];
        #pragma unroll
        for (int c = 0; c < 16; ++c) {
            int k = 4 * c + 2 * hi;
            bfrag[c].x = arow[k];
            bfrag[c].y = arow[k + 1];
        }
        float a2 = A2[r];
        float rs = Rsq[r];

        // 4-way split accumulators: dependency depth 4 instead of 16.
        v8f accv[4];
        #pragma unroll
        for (int j = 0; j < 4; ++j)
            accv[j] = (v8f){0.f, 0.f, 0.f, 0.f, 0.f, 0.f, 0.f, 0.f};
        #pragma unroll
        for (int c = 0; c < 16; ++c) {
            accv[c & 3] = __builtin_amdgcn_wmma_f32_16x16x4_f32(
                /*neg_a=*/false, afrag[c], /*neg_b=*/false, bfrag[c],
                /*c_mod=*/(short)0, accv[c & 3],
                /*reuse_a=*/false, /*reuse_b=*/false);
        }
        v8f acc = (accv[0] + accv[1]) + (accv[2] + accv[3]);

        // e[b,r] = exp((2*dot - A2[r])*invO2) * R[r]^2   (exp(-X2) cancels)
        #pragma unroll
        for (int i = 0; i < 8; ++i) {
            int m = i + 8 * hi;                        // batch row in tile
            float e = expf((2.0f * acc[i] - a2) * invO2) * rs;
            lds_e[m * RULES + r] = e;
        }
    }
    __syncthreads();

    // --- per-row normalization sum ----------------------------------------
    {
        int m = tid >> 4, j = tid & 15;
        float s = 0.f;
        #pragma unroll 8
        for (int i = 0; i < RULES / 16; ++i) s += lds_e[m * RULES + j + 16 * i];
        lds_part[m * 16 + j] = s;
    }
    __syncthreads();
    if (tid < 16) {
        float s = 0.f;
        #pragma unroll
        for (int j = 0; j < 16; ++j) s += lds_part[tid * 16 + j];
        lds_inv[tid] = 1.0f / s;
    }
    __syncthreads();

    // --- ER product combination: 9 columns (8 beta cols + q with beta=0) --
    if (tid < 16 * 9) {
        int b  = tid / 9;
        int kk = tid - b * 9;
        float inv  = lds_inv[b];
        float prod = 1.0f;
        #pragma unroll 8
        for (int r = 0; r < RULES; ++r) {
            float w  = lds_e[b * RULES + r] * inv;
            float bv = (kk < RES) ? lds_beta[r * RES + kk] : 0.0f;
            prod *= fmaf(w, bv - 1.0f, 1.0f);          // w*beta + (1 - w)
        }
        lds_p[b * 9 + kk] = prod;
    }
    __syncthreads();

    // --- belief + utility sum ---------------------------------------------
    if (tid < 16) {
        float p[RES];
        float sump = 0.f;
        #pragma unroll
        for (int k = 0; k < RES; ++k) { p[k] = lds_p[tid * 9 + k]; sump += p[k]; }
        float q     = lds_p[tid * 9 + RES];
        float denom = sump - (float)RES * q;
        float o = 0.f;
        #pragma unroll
        for (int k = 0; k < RES; ++k) o += (p[k] - q) / denom * U[k];
        out[b0 + tid] = o;
    }
}

// ---------------------------------------------------------------------------
extern "C" void kernel_launch(void* const* d_in, const int* in_sizes, int n_in,
                              void* d_out, int out_size, void* d_ws, size_t ws_size,
                              hipStream_t stream) {
    const float* x  = (const float*)d_in[0];
    const float* Ar = (const float*)d_in[1];
    const float* E  = (const float*)d_in[2];
    const float* O  = (const float*)d_in[3];
    const float* B  = (const float*)d_in[4];
    const float* R  = (const float*)d_in[5];
    const float* U  = (const float*)d_in[6];
    float* out = (float*)d_out;

    float* ws   = (float*)d_ws;
    float* attw = ws;                     // 64
    float* A2   = ws + 64;                // 1024
    float* Rsq  = ws + 64 + 1024;         // 1024
    float* beta = ws + 64 + 2048;         // 1024*8

    prep_attw <<<1, 64, 0, stream>>>(E, attw);
    prep_rules<<<RULES / 128, 128, 0, stream>>>(Ar, B, R, attw, A2, Rsq, beta);

    size_t smem = (size_t)(16 * RULES + RULES * RES + 16 * 16 + 16 + 16 * 9)
                  * sizeof(float);
    brb_main  <<<BATCH / 16, 256, smem, stream>>>(x, Ar, O, U, attw, A2, Rsq,
                                                  beta, out);
}